// RoIPool_515396075755
// MI455X (gfx1250) — compile-verified
//
#include <hip/hip_runtime.h>
#include <hip/hip_bf16.h>
#include <math.h>

// RoIPool (torchvision semantics), sizes fixed by the reference:
//   feats: [B=4, C=256, H=50, W=50] f32
//   rois:  [N, 5] f32  (batch_idx, x1, y1, x2, y2), N = in_sizes[1]/5
//   out:   [N, C, 7, 7] f32
//
// CDNA5 strategy: the feature map (10 MB) is L2-resident; the kernel is
// L2-request bound. Stage each ROI window into LDS with the gfx1250
// async global->LDS copy path (coalesced along w, ASYNCcnt-tracked), then
// do the irregular per-bin max reduction from LDS, and emit coalesced
// consecutive-float stores.

namespace {
constexpr int C    = 256;
constexpr int H    = 50;
constexpr int W    = 50;
constexpr int P    = 7;
constexpr int BINS = P * P;              // 49
constexpr int CH   = 4;                  // channels staged per block
constexpr int CSTRIDE = H * (W + 1);     // fixed per-channel LDS stride (2550 floats)
constexpr float SCALE = 0.0625f;         // 1/16
}

typedef __attribute__((address_space(1))) int gint;   // global int
typedef __attribute__((address_space(3))) int lint;   // LDS int

__global__ __launch_bounds__(256) void roipool_async_kernel(
    const float* __restrict__ feats,
    const float* __restrict__ rois,
    float* __restrict__ out)
{
    const int n  = blockIdx.x;               // ROI index
    const int c0 = blockIdx.y * CH;          // first channel of this block
    const int t  = threadIdx.x;              // 0..255

    __shared__ float stage[CH * CSTRIDE];    // 4 * 2550 * 4B = 40.8 KB (WGP has 320 KB)

    // ---- ROI parameters (uniform; bit-exact f32 replication of the reference)
    const float* r = rois + (size_t)n * 5;
    const int   b  = (int)r[0];
    const float x1 = rintf(r[1] * SCALE);    // jnp.round == RNE == rintf
    const float y1 = rintf(r[2] * SCALE);
    const float x2 = rintf(r[3] * SCALE);
    const float y2 = rintf(r[4] * SCALE);
    const float bw = fmaxf(x2 - x1 + 1.0f, 1.0f) * (1.0f / 7.0f);
    const float bh = fmaxf(y2 - y1 + 1.0f, 1.0f) * (1.0f / 7.0f);

    // Full ROI window (union of all bins), clipped to the feature map.
    const int hs0 = (int)fminf(fmaxf(y1, 0.0f), (float)H);
    const int heN = (int)fminf(fmaxf(ceilf(7.0f * bh) + y1, 0.0f), (float)H);
    const int ws0 = (int)fminf(fmaxf(x1, 0.0f), (float)W);
    const int weN = (int)fminf(fmaxf(ceilf(7.0f * bw) + x1, 0.0f), (float)W);
    const int winH = heN - hs0;
    const int winW = weN - ws0;
    const int winWp = winW | 1;              // odd LDS row stride -> spread banks

    // ---- Async stage: copy CH channel windows into LDS, coalesced along w.
    // Consecutive lanes take consecutive elements -> consecutive global
    // addresses -> full-line L2 requests instead of 32 scattered ones.
    const int rowTotal = winH * winW;
    const int total    = CH * rowTotal;
    const float* gbase = feats + ((size_t)b * C + c0) * (size_t)(H * W)
                               + (size_t)hs0 * W + ws0;
    for (int e = t; e < total; e += 256) {
        const int cl  = e / rowTotal;
        const int rem = e - cl * rowTotal;
        const int h   = rem / winW;
        const int w   = rem - h * winW;
        const float* gp = gbase + (size_t)cl * (H * W) + h * W + w;
        float*       lp = &stage[cl * CSTRIDE + h * winWp + w];
        __builtin_amdgcn_global_load_async_to_lds_b32((gint*)gp, (lint*)lp,
                                                      /*offset=*/0, /*cpol=*/0);
    }
    // Wait for this wave's async copies (ASYNCcnt), then sync all waves.
    asm volatile("s_wait_asynccnt 0x0" ::: "memory");
    __syncthreads();

    // ---- Compute: thread t -> (channel c_local, bin). 196 active threads;
    // consecutive threads store consecutive floats of out (coalesced).
    if (t < CH * BINS) {
        const int cl  = t / BINS;
        const int bin = t - cl * BINS;
        const int ph  = bin / 7;
        const int pw  = bin - ph * 7;

        // Bin window, identical f32 op sequence as the reference.
        const int hs = (int)fminf(fmaxf(floorf((float)ph * bh) + y1, 0.0f), (float)H);
        const int he = (int)fminf(fmaxf(ceilf((float)(ph + 1) * bh) + y1, 0.0f), (float)H);
        const int ws = (int)fminf(fmaxf(floorf((float)pw * bw) + x1, 0.0f), (float)W);
        const int we = (int)fminf(fmaxf(ceilf((float)(pw + 1) * bw) + x1, 0.0f), (float)W);

        float v;
        if (hs >= he || ws >= we) {
            v = 0.0f;                        // empty bin -> 0 (torchvision)
        } else {
            v = -INFINITY;
            const float* sp = &stage[cl * CSTRIDE];
            for (int h = hs; h < he; ++h) {
                const float* srow = sp + (h - hs0) * winWp - ws0;
                for (int w = ws; w < we; ++w)
                    v = fmaxf(v, srow[w]);
            }
        }
        out[((size_t)n * C + c0 + cl) * BINS + bin] = v;
    }
}

extern "C" void kernel_launch(void* const* d_in, const int* in_sizes, int n_in,
                              void* d_out, int out_size, void* d_ws, size_t ws_size,
                              hipStream_t stream) {
    (void)n_in; (void)out_size; (void)d_ws; (void)ws_size;
    const float* feats = (const float*)d_in[0];
    const float* rois  = (const float*)d_in[1];
    float*       out   = (float*)d_out;

    const int N = in_sizes[1] / 5;           // rois: [N, 5]
    dim3 grid((unsigned)N, C / CH);          // 256 x 64 blocks, 256 threads each
    roipool_async_kernel<<<grid, 256, 0, stream>>>(feats, rois, out);
}